// InfoNCELoss_8976481649056
// MI455X (gfx1250) — compile-verified
//
#include <hip/hip_runtime.h>
#include <math.h>

typedef float v2f __attribute__((ext_vector_type(2)));
typedef float v8f __attribute__((ext_vector_type(8)));

#define BHW      8192
#define DDIM     128
#define KNEG     100
#define TEMP     0.07f
#define EPSN     1e-12f
#define TIECAP   64

// ---------------------------------------------------------------------------
// prep: transpose (b,d,h,w) -> flat (bhw, d), row squared norms, positive term
// ---------------------------------------------------------------------------
__global__ void prep_kernel(const float* __restrict__ x1, const float* __restrict__ x2,
                            float* __restrict__ flatX, float* __restrict__ xsq,
                            float* __restrict__ pos) {
  int row = blockIdx.x;        // 0..8191 = b*1024 + h*32 + w
  int dd  = threadIdx.x;       // 0..127
  int b   = row >> 10;
  int hw  = row & 1023;
  float v1 = x1[(size_t)(b * DDIM + dd) * 1024 + hw];
  float v2 = x2[(size_t)(b * DDIM + dd) * 1024 + hw];
  flatX[(size_t)row * DDIM + dd] = v1;

  __shared__ float s1[DDIM];
  __shared__ float s2[DDIM];
  s1[dd] = v1 * v1;
  s2[dd] = v2 * v2;
  __syncthreads();
  for (int off = DDIM / 2; off > 0; off >>= 1) {
    if (dd < off) { s1[dd] += s1[dd + off]; s2[dd] += s2[dd + off]; }
    __syncthreads();
  }
  float sum1 = s1[0];
  float sum2 = s2[0];
  __syncthreads();

  float n1 = fmaxf(sqrtf(sum1), EPSN);
  float n2 = fmaxf(sqrtf(sum2), EPSN);
  float e  = expf(((v1 / n1) * (v2 / n2)) / TEMP);
  s1[dd] = e;
  __syncthreads();
  for (int off = DDIM / 2; off > 0; off >>= 1) {
    if (dd < off) s1[dd] += s1[dd + off];
    __syncthreads();
  }
  if (dd == 0) {
    pos[row] = s1[0];
    xsq[row] = sum1;
  }
}

// ---------------------------------------------------------------------------
// gemm: G[m - chunk_base][n] = dot(X[m], X[n]) via V_WMMA_F32_16X16X4_F32.
// Register-blocked: each wave computes 2 M-tiles x 4 N-tiles (8 accumulators),
// so a k-step issues 6 b64 loads for 8 WMMAs (10.7 FLOP/B at the vmem path).
// Workgroup: 256 threads = 8 waves; WG tile = 32 rows x 512 cols.
// ---------------------------------------------------------------------------
__global__ void gemm_kernel(const float* __restrict__ X, float* __restrict__ G,
                            int chunk_base) {
  int lane = threadIdx.x & 31;
  int wave = threadIdx.x >> 5;                // 0..7
  int m0_local = blockIdx.y * 32;             // 2 M tiles of 16
  int m0 = chunk_base + m0_local;
  int n0 = blockIdx.x * 512 + wave * 64;      // 4 N tiles of 16

  int lm = lane & 15;
  int kh = (lane >> 4) * 2;                   // K sub-pair per half-wave

  const float* __restrict__ arow0 = X + (size_t)(m0 + lm) * DDIM + kh;
  const float* __restrict__ arow1 = arow0 + (size_t)16 * DDIM;
  const float* __restrict__ brow0 = X + (size_t)(n0 + lm) * DDIM + kh;
  const float* __restrict__ brow1 = brow0 + (size_t)16 * DDIM;
  const float* __restrict__ brow2 = brow0 + (size_t)32 * DDIM;
  const float* __restrict__ brow3 = brow0 + (size_t)48 * DDIM;

  v8f acc[2][4];
#pragma unroll
  for (int m = 0; m < 2; ++m)
#pragma unroll
    for (int t = 0; t < 4; ++t) acc[m][t] = (v8f){0.f,0.f,0.f,0.f,0.f,0.f,0.f,0.f};

#pragma unroll
  for (int k = 0; k < DDIM; k += 4) {
    v2f a0 = *(const v2f*)(arow0 + k);
    v2f a1 = *(const v2f*)(arow1 + k);
    v2f b0 = *(const v2f*)(brow0 + k);
    v2f b1 = *(const v2f*)(brow1 + k);
    v2f b2 = *(const v2f*)(brow2 + k);
    v2f b3 = *(const v2f*)(brow3 + k);
    acc[0][0] = __builtin_amdgcn_wmma_f32_16x16x4_f32(false, a0, false, b0, (short)0, acc[0][0], false, false);
    acc[0][1] = __builtin_amdgcn_wmma_f32_16x16x4_f32(false, a0, false, b1, (short)0, acc[0][1], false, false);
    acc[0][2] = __builtin_amdgcn_wmma_f32_16x16x4_f32(false, a0, false, b2, (short)0, acc[0][2], false, false);
    acc[0][3] = __builtin_amdgcn_wmma_f32_16x16x4_f32(false, a0, false, b3, (short)0, acc[0][3], false, false);
    acc[1][0] = __builtin_amdgcn_wmma_f32_16x16x4_f32(false, a1, false, b0, (short)0, acc[1][0], false, false);
    acc[1][1] = __builtin_amdgcn_wmma_f32_16x16x4_f32(false, a1, false, b1, (short)0, acc[1][1], false, false);
    acc[1][2] = __builtin_amdgcn_wmma_f32_16x16x4_f32(false, a1, false, b2, (short)0, acc[1][2], false, false);
    acc[1][3] = __builtin_amdgcn_wmma_f32_16x16x4_f32(false, a1, false, b3, (short)0, acc[1][3], false, false);
  }

  // C layout: VGPR r -> row tileM0 + r + 8*(lane>>4), col tileN0 + (lane&15)
  int rsub = (lane >> 4) * 8;
#pragma unroll
  for (int m = 0; m < 2; ++m) {
    int rbase = m0_local + m * 16 + rsub;
#pragma unroll
    for (int t = 0; t < 4; ++t) {
      int col = n0 + t * 16 + lm;
#pragma unroll
      for (int r = 0; r < 8; ++r) {
        G[(size_t)(rbase + r) * BHW + col] = acc[m][t][r];
      }
    }
  }
}

// ---------------------------------------------------------------------------
// select: per row, radix-select the top-100 largest d2' = xsq[j] - 2*G
// (xsq[i] constant offset dropped), then accumulate exp(G/(ni*nj)/TEMP).
// 4 histogram passes on bit-flipped float keys (256 bins <-> 256 threads),
// 1 accumulation pass, deterministic smallest-index tie fixup.
// ---------------------------------------------------------------------------
__device__ __forceinline__ unsigned flip_key(float f) {
  unsigned u = __float_as_uint(f);
  unsigned m = (unsigned)((int)u >> 31) | 0x80000000u;  // neg->~, pos->flip sign
  return u ^ m;
}
__device__ __forceinline__ float unflip_key(unsigned k) {
  unsigned u = (k & 0x80000000u) ? (k ^ 0x80000000u) : ~k;
  return __uint_as_float(u);
}

__global__ void select_kernel(const float* __restrict__ G, const float* __restrict__ xsq,
                              float* __restrict__ negsum, int chunk_base) {
  int rl  = blockIdx.x;
  int i   = chunk_base + rl;
  int tid = threadIdx.x;           // 256 threads

  __shared__ unsigned fk[BHW];     // 32 KB flipped keys
  __shared__ unsigned hist[256];
  __shared__ float    sred[256];
  __shared__ unsigned sh_bin, sh_kr, sh_tiecnt;
  __shared__ unsigned tieidx[TIECAP];
  __shared__ float    sh_tiesum;

  const float* __restrict__ grow = G + (size_t)rl * BHW;
  for (int j = tid; j < BHW; j += 256) {
    float g  = __builtin_nontemporal_load(grow + j);   // single-use, L2-resident
    float d2 = xsq[j] - 2.0f * g;
    fk[j] = flip_key(d2);
  }
  if (tid == 0) { sh_tiecnt = 0; sh_tiesum = 0.0f; }
  __syncthreads();

  float ni = fmaxf(sqrtf(xsq[i]), EPSN);

  // --- radix select: find threshold key T of the 100th largest ---
  unsigned prefix = 0, pmask = 0, kr = KNEG;
  for (int r = 3; r >= 0; --r) {
    int shift = r * 8;
    hist[tid] = 0;
    __syncthreads();
    for (int j = tid; j < BHW; j += 256) {
      unsigned key = fk[j];
      if ((key & pmask) == prefix) atomicAdd(&hist[(key >> shift) & 255u], 1u);
    }
    __syncthreads();
    if (tid == 0) {
      unsigned cum = 0, bin = 0, krn = kr;
      for (int b2 = 255; b2 >= 0; --b2) {
        cum += hist[b2];
        if (cum >= kr) { bin = (unsigned)b2; krn = kr - (cum - hist[b2]); break; }
      }
      sh_bin = bin;
      sh_kr  = krn;
    }
    __syncthreads();
    prefix |= (sh_bin << shift);
    pmask  |= (0xFFu << shift);
    kr      = sh_kr;                 // #keys equal to prefix still needed
    __syncthreads();
  }
  unsigned T = prefix;               // exact key of the 100th largest

  // --- accumulate exp over keys strictly above T; collect ties ---
  float acc = 0.0f;
  for (int j = tid; j < BHW; j += 256) {
    unsigned key = fk[j];
    if (key > T) {
      float d2v = unflip_key(key);
      float gij = 0.5f * (xsq[j] - d2v);
      float nj  = fmaxf(sqrtf(xsq[j]), EPSN);
      acc += expf(gij / (ni * nj) / TEMP);
    } else if (key == T) {
      unsigned p = atomicAdd(&sh_tiecnt, 1u);
      if (p < TIECAP) tieidx[p] = (unsigned)j;
    }
  }
  sred[tid] = acc;
  __syncthreads();

  if (tid == 0) {
    // deterministic tie handling: take the kr smallest indices among ties
    unsigned n = sh_tiecnt < TIECAP ? sh_tiecnt : TIECAP;
    for (unsigned a = 1; a < n; ++a) {           // insertion sort (tiny n)
      unsigned v = tieidx[a]; int b2 = (int)a - 1;
      while (b2 >= 0 && tieidx[b2] > v) { tieidx[b2 + 1] = tieidx[b2]; --b2; }
      tieidx[b2 + 1] = v;
    }
    unsigned take = kr < n ? kr : n;
    float ts = 0.0f;
    float d2v = unflip_key(T);
    for (unsigned a = 0; a < take; ++a) {
      unsigned j = tieidx[a];
      float gij = 0.5f * (xsq[j] - d2v);
      float nj  = fmaxf(sqrtf(xsq[j]), EPSN);
      ts += expf(gij / (ni * nj) / TEMP);
    }
    sh_tiesum = ts;
  }
  __syncthreads();

  for (int off = 128; off > 0; off >>= 1) {
    if (tid < off) sred[tid] += sred[tid + off];
    __syncthreads();
  }
  if (tid == 0) negsum[i] = sred[0] + sh_tiesum;
}

// ---------------------------------------------------------------------------
// finalize: mean over rows of  log(pos+neg) - log(pos)
// ---------------------------------------------------------------------------
__global__ void finalize_kernel(const float* __restrict__ pos,
                                const float* __restrict__ neg,
                                float* __restrict__ out) {
  int tid = threadIdx.x;
  __shared__ float s[256];
  float a = 0.0f;
  for (int i = tid; i < BHW; i += 256) {
    float p = pos[i];
    float n = neg[i];
    a += (logf(p + n) - logf(p));
  }
  s[tid] = a;
  __syncthreads();
  for (int off = 128; off > 0; off >>= 1) {
    if (tid < off) s[tid] += s[tid + off];
    __syncthreads();
  }
  if (tid == 0) out[0] = s[0] / (float)BHW;
}

extern "C" void kernel_launch(void* const* d_in, const int* in_sizes, int n_in,
                              void* d_out, int out_size, void* d_ws, size_t ws_size,
                              hipStream_t stream) {
  (void)in_sizes; (void)n_in; (void)out_size;
  const float* x1 = (const float*)d_in[0];
  const float* x2 = (const float*)d_in[1];

  float* ws    = (float*)d_ws;
  float* flatX = ws;                              // 8192*128 f32 = 4 MB
  float* xsq   = flatX + (size_t)BHW * DDIM;      // 32 KB
  float* pos   = xsq + BHW;                       // 32 KB
  float* neg   = pos + BHW;                       // 32 KB
  float* G     = neg + BHW;                       // chunked Gram block

  size_t used  = (size_t)((char*)G - (char*)ws);
  size_t avail = ws_size > used ? ws_size - used : 0;
  long maxR = (long)(avail / ((size_t)BHW * sizeof(float)));
  int R = (int)(maxR > BHW ? BHW : maxR);
  R &= ~31;                       // WG covers 32 rows
  if (R < 32)   R = 32;           // minimal chunk (assume ws covers this)
  if (R > 2048) R = 2048;         // cap G block at 64 MB -> stays L2-resident

  prep_kernel<<<BHW, DDIM, 0, stream>>>(x1, x2, flatX, xsq, pos);

  for (int base = 0; base < BHW; base += R) {
    int rows = (BHW - base < R) ? (BHW - base) : R;   // multiple of 32
    dim3 grid(BHW / 512, rows / 32);
    gemm_kernel<<<grid, 256, 0, stream>>>(flatX, G, base);
    select_kernel<<<rows, 256, 0, stream>>>(G, xsq, neg, base);
  }

  finalize_kernel<<<1, 256, 0, stream>>>(pos, neg, (float*)d_out);
}